// Timewarp_68874095559123
// MI455X (gfx1250) — compile-verified
//
#include <hip/hip_runtime.h>
#include <stdint.h>
#include <math.h>

// Problem constants (from reference): B=64, C=128, T=4096, KNOT=4
#define Bn 64
#define Cn 128
#define Tn 4096
#define NK 6                 // KNOT + 2
#define HH 819.0f            // h = (T-1)/(KNOT+1) = 4095/5

typedef __attribute__((ext_vector_type(2))) float v2f;
typedef __attribute__((ext_vector_type(8))) float v8f;

// ---------------------------------------------------------------------------
// Kernel 0: build the 6x6 not-a-knot matrix A and invert it (Gauss-Jordan,
// partial pivoting, single thread -- negligible cost). Store zero-padded as
// a 16x8 row-major tile so the WMMA kernel can load fragments with uniform,
// in-bounds addressing (keeps EXEC uniform around the WMMA).
// ---------------------------------------------------------------------------
__global__ void tw_build_ainv(float* __restrict__ ainv_pad) {
    if (threadIdx.x != 0 || blockIdx.x != 0) return;
    float A[NK][2 * NK];
    for (int i = 0; i < NK; ++i)
        for (int j = 0; j < 2 * NK; ++j) A[i][j] = 0.0f;
    const float h = HH;
    for (int i = 1; i < NK - 1; ++i) {
        A[i][i - 1] = h / 6.0f;
        A[i][i]     = 2.0f * h / 3.0f;
        A[i][i + 1] = h / 6.0f;
    }
    A[0][0] = 1.0f; A[0][1] = -2.0f; A[0][2] = 1.0f;
    A[NK - 1][NK - 3] = 1.0f; A[NK - 1][NK - 2] = -2.0f; A[NK - 1][NK - 1] = 1.0f;
    for (int i = 0; i < NK; ++i) A[i][NK + i] = 1.0f;

    for (int col = 0; col < NK; ++col) {
        int piv = col; float best = fabsf(A[col][col]);
        for (int r = col + 1; r < NK; ++r) {
            float v = fabsf(A[r][col]);
            if (v > best) { best = v; piv = r; }
        }
        if (piv != col)
            for (int j = 0; j < 2 * NK; ++j) {
                float t = A[col][j]; A[col][j] = A[piv][j]; A[piv][j] = t;
            }
        float inv = 1.0f / A[col][col];
        for (int j = 0; j < 2 * NK; ++j) A[col][j] *= inv;
        for (int r = 0; r < NK; ++r) {
            if (r == col) continue;
            float f = A[r][col];
            for (int j = 0; j < 2 * NK; ++j) A[r][j] -= f * A[col][j];
        }
    }
    // padded 16x8 tile (rows >=6 and cols >=6 are zero)
    for (int i = 0; i < 16; ++i)
        for (int j = 0; j < 8; ++j)
            ainv_pad[i * 8 + j] = (i < NK && j < NK) ? A[i][NK + j] : 0.0f;
}

// rhs(b,k,c): interior rows = second difference of yy / h, boundary rows = 0.
// k >= 5 also returns 0, which doubles as the K-dim zero padding (K=6 -> 8).
__device__ __forceinline__ float rhs_val(const float* __restrict__ yy,
                                         int b, int k, int c, float invh) {
    if (k <= 0 || k >= NK - 1) return 0.0f;
    const float* p = yy + (size_t)(b * NK + k) * Cn + c;
    return (p[Cn] - 2.0f * p[0] + p[-Cn]) * invh;
}

// ---------------------------------------------------------------------------
// Kernel 1: batched spline solve M[b] = Ainv(6x6) @ rhs[b](6x128) using
// V_WMMA_F32_16X16X4_F32. One wave per 16x16 output tile; 512 tiles total
// (64 batches x 8 column tiles), K padded 6 -> 8 (two K=4 WMMA steps).
// ---------------------------------------------------------------------------
__global__ __launch_bounds__(128)
void tw_solve_wmma(const float* __restrict__ yy,
                   const float* __restrict__ ainv_pad,
                   float* __restrict__ Mout) {
    const int wave = threadIdx.x >> 5;
    const int lane = threadIdx.x & 31;
    const int tile = blockIdx.x * 4 + wave;   // 0..511
    const int b    = tile >> 3;               // batch
    const int n0   = (tile & 7) << 4;         // column tile base in C
    const int half = lane >> 4;               // lane group selects K pair
    const int m    = lane & 15;               // row (A) / column (B) within tile
    const float invh = 1.0f / HH;

    v8f acc = {0.0f, 0.0f, 0.0f, 0.0f, 0.0f, 0.0f, 0.0f, 0.0f};
#pragma unroll
    for (int s = 0; s < 2; ++s) {
        const int ka = s * 4 + half * 2;      // K index of this lane's VGPR0 element
        v2f a, bb;
        a.x = ainv_pad[m * 8 + ka];
        a.y = ainv_pad[m * 8 + ka + 1];
        const int c = n0 + m;
        bb.x = rhs_val(yy, b, ka,     c, invh);
        bb.y = rhs_val(yy, b, ka + 1, c, invh);
        acc = __builtin_amdgcn_wmma_f32_16x16x4_f32(
            false, a, false, bb, (short)0, acc, false, false);
    }
    // C/D layout: VGPR r, lanes 0-15 => M=r, N=lane. Rows 0..5 are valid.
    if (lane < 16) {
        const int c = n0 + lane;
        Mout[(size_t)(b * NK + 0) * Cn + c] = acc[0];
        Mout[(size_t)(b * NK + 1) * Cn + c] = acc[1];
        Mout[(size_t)(b * NK + 2) * Cn + c] = acc[2];
        Mout[(size_t)(b * NK + 3) * Cn + c] = acc[3];
        Mout[(size_t)(b * NK + 4) * Cn + c] = acc[4];
        Mout[(size_t)(b * NK + 5) * Cn + c] = acc[5];
    }
}

// ---------------------------------------------------------------------------
// Kernel 2: per-row timewarp. One 256-thread workgroup (8 wave32) per (b,c)
// row. The 16KB x-row is copied global->LDS with CDNA5 async-LDS b128 loads,
// issued at kernel entry so the DMA overlaps the entire spline-curve +
// block-scan phase (ASYNCcnt tracked; s_wait_asynccnt + barrier before use).
// LDS: 2*16KB + scan scratch (well under 320KB/WGP -> several WGs resident).
// ---------------------------------------------------------------------------
__global__ __launch_bounds__(256)
void tw_warp(const float* __restrict__ x,
             const float* __restrict__ yy,
             const float* __restrict__ Msol,
             const float* __restrict__ mask_rand,
             float* __restrict__ out) {
    __shared__ float s_xp[Tn];
    __shared__ float s_fp[Tn];
    __shared__ float s_red[256];
    __shared__ float s_M[NK];
    __shared__ float s_P[NK];   // y/h - M*h/6

    const int row = blockIdx.x;          // 0 .. B*C-1
    const int b   = row / Cn;
    const int c   = row % Cn;
    const int tid = threadIdx.x;
    const float h = HH, invh = 1.0f / HH;

    const float* xrow = x + (size_t)row * Tn;

    // --- async global->LDS copy of the whole x row (4 x b128 per thread) ---
    // Flat pointers into LDS are {shared_aperture_hi32, lds_offset32}; the
    // low 32 bits are exactly the LDS byte address the async op consumes.
    {
        const unsigned lds0  = (unsigned)(unsigned long long)(uintptr_t)&s_fp[0];
        const unsigned long long gbase = (unsigned long long)(uintptr_t)xrow;
#pragma unroll
        for (int k = 0; k < 4; ++k) {
            const unsigned byteoff = (unsigned)((k * 256 + tid) * 16);
            asm volatile("global_load_async_to_lds_b128 %0, %1, %2"
                         :: "v"(lds0 + byteoff), "v"(byteoff), "s"(gbase)
                         : "memory");
        }
    }

    if (tid < NK) {
        float Mi = Msol[(size_t)(b * NK + tid) * Cn + c];
        float yi = yy  [(size_t)(b * NK + tid) * Cn + c];
        s_M[tid] = Mi;
        s_P[tid] = yi * invh - Mi * (h / 6.0f);
    }
    __syncthreads();

    // --- spline curve + local inclusive prefix over 16 contiguous samples ---
    float loc[16];
    float run = 0.0f;
    const int t0 = tid * 16;
#pragma unroll
    for (int j = 0; j < 16; ++j) {
        float t   = (float)(t0 + j);
        int  idx  = (int)floorf(t / h);
        idx = idx > NK - 2 ? NK - 2 : idx;
        float xl  = (float)idx * h;
        float dxl = t - xl;
        float dxr = xl + h - t;
        float Mi  = s_M[idx], Mi1 = s_M[idx + 1];
        float Pi  = s_P[idx], Pi1 = s_P[idx + 1];
        float cur = (Mi * dxr * dxr * dxr + Mi1 * dxl * dxl * dxl) * (invh * (1.0f / 6.0f))
                  + Pi * dxr + Pi1 * dxl;
        run += cur;
        loc[j] = run;
    }
    s_red[tid] = run;
    __syncthreads();

    // --- Hillis-Steele inclusive scan of the 256 thread totals (wave32-safe) ---
#pragma unroll
    for (int off = 1; off < 256; off <<= 1) {
        float v   = s_red[tid];
        float add = (tid >= off) ? s_red[tid - off] : 0.0f;
        __syncthreads();
        s_red[tid] = v + add;
        __syncthreads();
    }
    const float excl  = (tid == 0) ? 0.0f : s_red[tid - 1];
    const float total = s_red[255];
    const float scale = (float)(Tn - 1) / total;
#pragma unroll
    for (int j = 0; j < 16; ++j)
        s_xp[t0 + j] = (excl + loc[j]) * scale;

    // --- drain the async row copy, make it block-visible ---
    asm volatile("s_wait_asynccnt 0" ::: "memory");
    __syncthreads();

    // --- inverse-warp via binary-search interp (jnp.interp semantics) ---
    const bool use_warp = mask_rand[b] < 0.5f;
    float* orow = out + (size_t)row * Tn;
    const float xp0 = s_xp[0];
    const float xpl = s_xp[Tn - 1];
#pragma unroll 4
    for (int k = 0; k < 16; ++k) {
        const int q  = k * 256 + tid;
        const float tq = (float)q;
        float res;
        if (tq <= xp0) {
            res = s_fp[0];
        } else if (tq >= xpl) {
            res = s_fp[Tn - 1];
        } else {
            int lo = 0, hi = Tn - 1;          // invariant: xp[lo] <= tq < xp[hi]
            while (hi - lo > 1) {
                int mid = (lo + hi) >> 1;
                if (s_xp[mid] <= tq) lo = mid; else hi = mid;
            }
            float xa = s_xp[lo], xb = s_xp[lo + 1];
            float fa = s_fp[lo], fb = s_fp[lo + 1];
            res = fa + (fb - fa) * ((tq - xa) / (xb - xa));
        }
        // streaming store: written once, never re-read -> non-temporal
        __builtin_nontemporal_store(use_warp ? res : s_fp[q], &orow[q]);
    }
}

// ---------------------------------------------------------------------------
// Launch: inputs are x (B,C,T) f32, yy (B,6,C) f32, mask_rand (B,1,1) f32.
// Workspace layout: [0..127] padded Ainv tile, then B*6*C floats for M.
// ---------------------------------------------------------------------------
extern "C" void kernel_launch(void* const* d_in, const int* in_sizes, int n_in,
                              void* d_out, int out_size, void* d_ws, size_t ws_size,
                              hipStream_t stream) {
    const float* x         = (const float*)d_in[0];
    const float* yy        = (const float*)d_in[1];
    const float* mask_rand = (const float*)d_in[2];
    float* out  = (float*)d_out;
    float* ainv = (float*)d_ws;        // 128 floats (16x8 padded tile)
    float* Msol = ainv + 128;          // B*6*C floats

    tw_build_ainv<<<1, 32, 0, stream>>>(ainv);
    tw_solve_wmma<<<(Bn * Cn / 16) / 4, 128, 0, stream>>>(yy, ainv, Msol);
    tw_warp<<<Bn * Cn, 256, 0, stream>>>(x, yy, Msol, mask_rand, out);
}